// ContinuousConv1DSim_57621281243745
// MI455X (gfx1250) — compile-verified
//
#include <hip/hip_runtime.h>
#include <hip/hip_bf16.h>

// Problem constants (match reference)
#define Bb 32
#define Ll 2048
#define Cc 64
#define Oo 64
#define Ss 8
#define KS 9                 // KERNEL_SIZE+1 taps
#define TILE 32              // positions per block
#define ROWS 48              // 16 (pad, WMMA-aligned) + 32 positions
#define RSTRIDE 132          // LDS row stride in floats (bank-rotating pad)
#define OUTROWS ((Ll - 1) * KS + 1)   // 18424 rows per batch

typedef float v2f __attribute__((ext_vector_type(2)));
typedef float v4f __attribute__((ext_vector_type(4)));
typedef float v8f __attribute__((ext_vector_type(8)));

__global__ __launch_bounds__(256) void ccs_fused_kernel(
    const float* __restrict__ times,          // (B, L)
    const float* __restrict__ features,       // (B, L, C)
    const unsigned char* __restrict__ npm,    // (B, L) bool
    const float* __restrict__ usamp,          // (S,)
    const float* __restrict__ Wm,             // (O, C)
    const float* __restrict__ bp,             // (C, O)
    float* __restrict__ out)                  // (B, OUTROWS, O)
{
    __shared__ float sm_feat[ROWS * RSTRIDE]; // 48 rows x 128 cols (lin|bias), padded
    __shared__ float sm_time[ROWS + 4];
    __shared__ float sm_mask[ROWS + 4];
    __shared__ float sm_us[Ss];

    const int b  = blockIdx.y;
    const int l0 = blockIdx.x * TILE;
    const int tid = threadIdx.x;

    // ---- stage times / mask (LDS row q <-> global l = l0 - 16 + q) ----
    if (tid < ROWS + 1) {                    // need q = 0..48 (q=48 for udt)
        const int l = l0 - 16 + tid;
        const bool in = (l >= 0) && (l < Ll);
        sm_time[tid] = in ? times[(size_t)b * Ll + l] : 0.0f;
        sm_mask[tid] = (in && npm[(size_t)b * Ll + l]) ? 1.0f : 0.0f;
    }
    if (tid < Ss) sm_us[tid] = usamp[tid];

    // ---- WMMA phase: all_feats[q][0:128] = features row @ [W^T | bias_param] ----
    // wave w -> column tile nt = w (16 cols); row tiles rt = 0..2 (48 rows)
    {
        const int lane = tid & 31;
        const int nt   = tid >> 5;           // 0..7
        const int mrow = lane & 15;
        const int hi   = lane >> 4;          // lanes 16-31 hold K+2,K+3
        const int ncol = nt * 16 + (lane & 15);

        // Preload B operand for all 16 K-steps: Wc[k][ncol]
        v2f bop[16];
#pragma unroll
        for (int kk = 0; kk < 16; ++kk) {
            const int kb = kk * 4 + hi * 2;
            if (ncol < Oo) {                 // modified: W[ncol, k]
                bop[kk].x = Wm[(size_t)ncol * Cc + kb];
                bop[kk].y = Wm[(size_t)ncol * Cc + kb + 1];
            } else {                         // feat_bias: bp[k, ncol-64]
                bop[kk].x = bp[(size_t)kb * Oo + (ncol - Oo)];
                bop[kk].y = bp[(size_t)(kb + 1) * Oo + (ncol - Oo)];
            }
        }

#pragma unroll
        for (int rt = 0; rt < 3; ++rt) {
            const int q = rt * 16 + mrow;
            const int l = l0 - 16 + q;
            // Rows with l < 0 are only ever consumed under a 0-valued mask in
            // the window phase, so any finite value is fine: clamp the address
            // (branchless) instead of guarding/zeroing the loads.
            const int l_safe = (l < 0) ? 0 : l;
            const float* frow = features + ((size_t)b * Ll + l_safe) * Cc + hi * 2;

            // Hoist all 16 A-fragment loads: 16 b64 loads off one base, then
            // a single wait and an uninterrupted WMMA chain.
            v2f a[16];
#pragma unroll
            for (int kk = 0; kk < 16; ++kk)
                a[kk] = *(const v2f*)(frow + kk * 4);

            v8f acc0 = {0.f, 0.f, 0.f, 0.f, 0.f, 0.f, 0.f, 0.f};
            v8f acc1 = {0.f, 0.f, 0.f, 0.f, 0.f, 0.f, 0.f, 0.f};
#pragma unroll
            for (int kk = 0; kk < 8; ++kk) {
                acc0 = __builtin_amdgcn_wmma_f32_16x16x4_f32(
                    false, a[2 * kk],     false, bop[2 * kk],     (short)0, acc0, false, false);
                acc1 = __builtin_amdgcn_wmma_f32_16x16x4_f32(
                    false, a[2 * kk + 1], false, bop[2 * kk + 1], (short)0, acc1, false, false);
            }
            // D layout: VGPR i -> row M = i + 8*hi, col N = lane&15
#pragma unroll
            for (int i = 0; i < 8; ++i) {
                const int qq = rt * 16 + i + hi * 8;
                sm_feat[qq * RSTRIDE + nt * 16 + (lane & 15)] = acc0[i] + acc1[i];
            }
        }
    }
    __syncthreads();

    // ---- window + sample-expansion phase ----
    const int p  = tid >> 3;                 // 0..31 position in tile
    const int og = tid & 7;                  // 8-output group
    const int l  = l0 + p;
    const int qc = p + 16;

    const float tc = sm_time[qc];
    const float mc = sm_mask[qc];

    float sreal[8], ssim[8], ssl[8];
#pragma unroll
    for (int c = 0; c < 8; ++c) { sreal[c] = 0.f; ssim[c] = 0.f; ssl[c] = 0.f; }

#pragma unroll
    for (int j = 0; j <= 8; ++j) {
        const int q = qc - j;
        const float mj = mc * sm_mask[q];
        const float dt = mj * (tc - sm_time[q]);
        const float* fr = &sm_feat[q * RSTRIDE + og * 8];
#pragma unroll
        for (int c = 0; c < 8; ++c) {
            const float lv = fr[c];
            const float bv = fr[Oo + c];
            const float term = dt * lv + mj * bv;
            if (j >= 1) sreal[c] += term;            // real: lags 1..8
            if (j <= 7) { ssim[c] += term;           // sim : lags 0..7
                          ssl[c]  += mj * lv; }
        }
    }

    // udt[l] = m(l+1, j=1) * (t[l+1] - t[l])   (only used when l < L-1)
    const float mu  = mc * sm_mask[qc + 1];
    const float udt = mu * (sm_time[qc + 1] - tc);

    float* ob = out + ((size_t)b * OUTROWS + (size_t)l * KS) * Oo + og * 8;

    // row 0 of the group (or the single final row when l == L-1): real.
    // Output is a pure 151MB streaming write, never re-read: use NT stores.
    {
        v4f r0 = {sreal[0], sreal[1], sreal[2], sreal[3]};
        v4f r1 = {sreal[4], sreal[5], sreal[6], sreal[7]};
        __builtin_nontemporal_store(r0, (v4f*)(ob));
        __builtin_nontemporal_store(r1, (v4f*)(ob + 4));
    }
    if (l < Ll - 1) {
#pragma unroll
        for (int s = 0; s < Ss; ++s) {
            const float f = udt * sm_us[s];
            float* os = ob + (size_t)(1 + s) * Oo;
            v4f r0 = {ssim[0] + f * ssl[0], ssim[1] + f * ssl[1],
                      ssim[2] + f * ssl[2], ssim[3] + f * ssl[3]};
            v4f r1 = {ssim[4] + f * ssl[4], ssim[5] + f * ssl[5],
                      ssim[6] + f * ssl[6], ssim[7] + f * ssl[7]};
            __builtin_nontemporal_store(r0, (v4f*)(os));
            __builtin_nontemporal_store(r1, (v4f*)(os + 4));
        }
    }
}

extern "C" void kernel_launch(void* const* d_in, const int* in_sizes, int n_in,
                              void* d_out, int out_size, void* d_ws, size_t ws_size,
                              hipStream_t stream) {
    (void)in_sizes; (void)n_in; (void)out_size; (void)d_ws; (void)ws_size;
    const float*         times    = (const float*)d_in[0];
    const float*         features = (const float*)d_in[1];
    const unsigned char* npm      = (const unsigned char*)d_in[2];
    const float*         usamp    = (const float*)d_in[3];
    const float*         Wm       = (const float*)d_in[4];
    const float*         bp       = (const float*)d_in[5];
    float*               out      = (float*)d_out;

    dim3 grid(Ll / TILE, Bb);   // (64, 32)
    ccs_fused_kernel<<<grid, 256, 0, stream>>>(times, features, npm, usamp, Wm, bp, out);
}